// LearnableDenseConnections_4887672783218
// MI455X (gfx1250) — compile-verified
//
#include <hip/hip_runtime.h>
#include <stdint.h>

// Problem constants (from the reference)
#define B_  2048
#define N_  512
#define C_  64
#define L_  2
#define O_  8192
#define LO_ (L_ * O_)            // 16384 output columns

// Clang ext vectors (HIP's float4/int4 are structs; builtins need real vectors)
typedef __attribute__((ext_vector_type(4))) float v4f;
typedef __attribute__((ext_vector_type(4))) int   v4i;

// ---------------------------------------------------------------------------
// Phase 1: sel[lo] = indices[argmax_c weights[c, lo], lo]
// weights laid out [C, L*O]; first-max-wins to match jnp.argmax tie-breaking.
// ---------------------------------------------------------------------------
__global__ __launch_bounds__(256) void ldc_argmax_sel(
    const float* __restrict__ weights,
    const int*   __restrict__ indices,
    int*         __restrict__ sel)
{
    const int t = blockIdx.x * 256 + threadIdx.x;
    if (t >= LO_) return;

    float best  = weights[t];
    int   bestc = 0;
#pragma unroll 8
    for (int c = 1; c < C_; ++c) {
        float w = weights[(size_t)c * LO_ + t];
        if (w > best) { best = w; bestc = c; }   // strict > keeps first occurrence
    }
    sel[t] = indices[(size_t)bestc * LO_ + t];
}

// ---------------------------------------------------------------------------
// Phase 2: out[b, lo] = x[b, sel[lo]]
// Block = 256 threads; handles BTILE=8 batch rows x LOTILE=2048 columns.
// x rows staged into LDS via async global->LDS B128 copies (ASYNCcnt path),
// gathered with ds_load_b32, written back as coalesced NT b128 stores.
// ---------------------------------------------------------------------------
#define BTILE  8
#define LOTILE 2048

__global__ __launch_bounds__(256) void ldc_gather(
    const float* __restrict__ x,
    const int*   __restrict__ sel,
    float*       __restrict__ out)
{
    __shared__ __align__(16) float sx[BTILE * N_];   // 8 rows * 2 KiB = 16 KiB

    const int tid    = threadIdx.x;
    const int b0     = blockIdx.y * BTILE;
    const int lobase = blockIdx.x * LOTILE;

    // ---- async-copy x[b0 .. b0+7, :] (one contiguous 16 KiB span) into LDS
    const char*    gsrc     = (const char*)(x + (size_t)b0 * N_);
    const uint32_t lds_base = (uint32_t)(uintptr_t)(&sx[0]);   // flat->LDS offset (addr[31:0])
#pragma unroll
    for (int p = 0; p < 4; ++p) {
        const uint32_t off  = (uint32_t)(p * 256 + tid) * 16u; // 256 lanes * 16 B = 4 KiB/pass
        const uint32_t ldsa = lds_base + off;
        const char*    g    = gsrc + off;
        asm volatile("global_load_async_to_lds_b128 %0, %1, off"
                     :
                     : "v"(ldsa), "v"(g)
                     : "memory");
    }
    asm volatile("s_wait_asynccnt 0x0" ::: "memory");  // my wave's async copies done
    __syncthreads();                                   // everyone's copies visible

    // ---- each thread owns 8 consecutive output columns
    const int  lo0 = lobase + tid * 8;
    const v4i  s0  = *(const v4i*)(sel + lo0);
    const v4i  s1  = *(const v4i*)(sel + lo0 + 4);

#pragma unroll
    for (int r = 0; r < BTILE; ++r) {
        const float* row = &sx[r * N_];
        v4f v0, v1;
        v0.x = row[s0.x]; v0.y = row[s0.y]; v0.z = row[s0.z]; v0.w = row[s0.w];
        v1.x = row[s1.x]; v1.y = row[s1.y]; v1.z = row[s1.z]; v1.w = row[s1.w];
        float* op = out + (size_t)(b0 + r) * LO_ + lo0;
        __builtin_nontemporal_store(v0, (v4f*)op);        // output is write-once:
        __builtin_nontemporal_store(v1, (v4f*)(op + 4));  // NT keeps x/sel hot in L2
    }
}

// ---------------------------------------------------------------------------
// Harness entry point
// ---------------------------------------------------------------------------
extern "C" void kernel_launch(void* const* d_in, const int* in_sizes, int n_in,
                              void* d_out, int out_size, void* d_ws, size_t ws_size,
                              hipStream_t stream)
{
    const float* x       = (const float*)d_in[0];   // [B, N]   float32
    const float* weights = (const float*)d_in[1];   // [C, L, O] float32
    const int*   indices = (const int*)d_in[2];     // [C, L, O] int (per harness convention)
    float*       out     = (float*)d_out;           // [B, L, O] float32
    int*         sel     = (int*)d_ws;              // LO_ ints = 64 KiB scratch

    ldc_argmax_sel<<<dim3(LO_ / 256), dim3(256), 0, stream>>>(weights, indices, sel);
    ldc_gather<<<dim3(LO_ / LOTILE, B_ / BTILE), dim3(256), 0, stream>>>(x, sel, out);
}